// GCNConv_3693671874793
// MI455X (gfx1250) — compile-verified
//
#include <hip/hip_runtime.h>

typedef __attribute__((ext_vector_type(2))) float v2f;
typedef __attribute__((ext_vector_type(8))) float v8f;

#define D 128
#define LDSW 136  // padded LDS row stride (floats): 2*136 % 64 == 16 -> the two
                  // 16-lane halves of a wave hit disjoint bank sets on B reads

// ---- Phase 1: degree (self loop contributes 1 to every node) ----
__global__ void k_deg_init(float* __restrict__ deg, int N) {
    int i = blockIdx.x * blockDim.x + threadIdx.x;
    if (i < N) deg[i] = 1.0f;
}

__global__ void k_deg_edges(const int* __restrict__ ei, float* __restrict__ deg, int E) {
    int e = blockIdx.x * blockDim.x + threadIdx.x;
    if (e < E) atomicAdd(&deg[ei[E + e]], 1.0f);  // col = edge_index[1]
}

__global__ void k_dinv(const float* __restrict__ deg, float* __restrict__ dinv, int N) {
    int i = blockIdx.x * blockDim.x + threadIdx.x;
    if (i < N) dinv[i] = rsqrtf(deg[i]);  // deg >= 1 always (self loops)
}

// ---- Phase 2: xl = x @ W + b using V_WMMA_F32_16X16X4_F32 ----
// One wave computes a 16x128 tile of xl (16 rows, all 8 column tiles),
// so x is read exactly once. W staged in LDS (68 KB, padded stride).
__global__ void k_gemm(const float* __restrict__ x, const float* __restrict__ W,
                       const float* __restrict__ bias, float* __restrict__ xl,
                       int N, int mTiles) {
    extern __shared__ float lw[];  // [D][LDSW]
    int tid = threadIdx.x;

    // Stage W (k-major) into LDS: coalesced global reads, conflict-free writes.
    for (int i = tid; i < D * D; i += 256) {
        int k = i >> 7;
        int n = i & (D - 1);
        lw[k * LDSW + n] = W[i];
    }
    __syncthreads();

    int wave  = tid >> 5;
    int lane  = tid & 31;
    int mTile = blockIdx.x * 8 + wave;
    if (mTile >= mTiles) return;  // wave-uniform: EXEC stays all-ones for WMMA

    int l    = lane & 15;
    int half = lane >> 4;

    // A-matrix fp32 16x4 layout: lanes 0-15 hold M=l, VGPR{0,1}=K{0,1};
    // lanes 16-31 hold M=l, VGPR{0,1}=K{2,3}.
    int arow = mTile * 16 + l;
    if (arow >= N) arow = N - 1;               // clamp (N=50000 is 16-aligned anyway)
    const float* xrow = x + (long long)arow * D;

    v8f acc[8];
    v8f zero = {0.f, 0.f, 0.f, 0.f, 0.f, 0.f, 0.f, 0.f};
#pragma unroll
    for (int nt = 0; nt < 8; ++nt) acc[nt] = zero;

    for (int k0 = 0; k0 < D; k0 += 4) {
        int kk = k0 + 2 * half;                     // even -> 8B aligned
        v2f a = *(const v2f*)(xrow + kk);           // K=kk, kk+1 for this half
#pragma unroll
        for (int nt = 0; nt < 8; ++nt) {
            int n = nt * 16 + l;
            v2f bv;
            bv.x = lw[kk * LDSW + n];               // B[K=kk  ][n]
            bv.y = lw[(kk + 1) * LDSW + n];         // B[K=kk+1][n]
            acc[nt] = __builtin_amdgcn_wmma_f32_16x16x4_f32(
                false, a, false, bv, (short)0, acc[nt], false, false);
        }
    }

    // C/D layout: VGPR r holds rows {r, r+8} for lane halves {0,1}, col = n.
    // Wave-uniform full-tile check: hot path has zero per-store EXEC juggling.
    float* base = xl + (long long)(mTile * 16 + 8 * half) * D + l;
    if (mTile * 16 + 16 <= N) {
#pragma unroll
        for (int nt = 0; nt < 8; ++nt) {
            float bb = bias[nt * 16 + l];
#pragma unroll
            for (int r = 0; r < 8; ++r) {
                base[(long long)r * D + nt * 16] = acc[nt][r] + bb;
            }
        }
    } else {
#pragma unroll
        for (int nt = 0; nt < 8; ++nt) {
            float bb = bias[nt * 16 + l];
#pragma unroll
            for (int r = 0; r < 8; ++r) {
                int row = mTile * 16 + r + 8 * half;
                if (row < N) xl[(long long)row * D + nt * 16 + l] = acc[nt][r] + bb;
            }
        }
    }
}

// ---- Phase 3: out[i] = dinv[i]^2 * xl[i]  (self-loop term; also inits d_out) ----
__global__ void k_self(const float* __restrict__ xl, const float* __restrict__ dinv,
                       float* __restrict__ out, int N) {
    int idx = blockIdx.x * blockDim.x + threadIdx.x;
    int i = idx >> 5, l = idx & 31;
    if (i >= N) return;
    float s = dinv[i];
    float nrm = s * s;
    float4 v = ((const float4*)(xl + (long long)i * D))[l];
    float4 m = make_float4(nrm * v.x, nrm * v.y, nrm * v.z, nrm * v.w);
    ((float4*)(out + (long long)i * D))[l] = m;
}

// ---- Phase 4: edge scatter. 32 lanes per edge, float4 gather + 4 fp32 atomics ----
__global__ void k_scatter(const int* __restrict__ ei, const float* __restrict__ xl,
                          const float* __restrict__ dinv, float* __restrict__ out, int E) {
    int idx = blockIdx.x * blockDim.x + threadIdx.x;
    int e = idx >> 5, l = idx & 31;     // e is wave-uniform -> scalarizable loads
    if (e >= E) return;
    int r = ei[e];                       // row = edge_index[0]
    int c = ei[E + e];                   // col = edge_index[1]
    float nrm = dinv[r] * dinv[c];
    float4 v = ((const float4*)(xl + (long long)c * D))[l];
    float* o = out + (long long)r * D + l * 4;
    atomicAdd(o + 0, nrm * v.x);
    atomicAdd(o + 1, nrm * v.y);
    atomicAdd(o + 2, nrm * v.z);
    atomicAdd(o + 3, nrm * v.w);
}

extern "C" void kernel_launch(void* const* d_in, const int* in_sizes, int n_in,
                              void* d_out, int out_size, void* d_ws, size_t ws_size,
                              hipStream_t stream) {
    const float* x    = (const float*)d_in[0];
    const int*   ei   = (const int*)d_in[1];   // edge_index, [2, E] flat
    const float* W    = (const float*)d_in[2];
    const float* bias = (const float*)d_in[3];
    float*       out  = (float*)d_out;

    int N = in_sizes[0] / D;
    int E = in_sizes[1] / 2;

    // Workspace layout: deg[N] | dinv[N] | xl[N*D]   (~26 MB)
    float* deg  = (float*)d_ws;
    float* dinv = deg + N;
    float* xl   = dinv + N;

    // 1) degrees (self loop = 1, then atomic count over col)
    k_deg_init <<<(N + 255) / 256, 256, 0, stream>>>(deg, N);
    k_deg_edges<<<(E + 255) / 256, 256, 0, stream>>>(ei, deg, E);
    k_dinv     <<<(N + 255) / 256, 256, 0, stream>>>(deg, dinv, N);

    // 2) xl = x @ W + b  (WMMA fp32; 8 waves/block, 1 M-tile each)
    int mTiles = (N + 15) / 16;
    k_gemm<<<(mTiles + 7) / 8, 256, D * LDSW * (int)sizeof(float), stream>>>(
        x, W, bias, xl, N, mTiles);

    // 3) self-loop term (initializes every element of d_out)
    long long selfT = (long long)N * 32;
    k_self<<<(int)((selfT + 255) / 256), 256, 0, stream>>>(xl, dinv, out, N);

    // 4) edge gather-multiply-scatter (L2-resident atomics)
    long long scatT = (long long)E * 32;
    k_scatter<<<(int)((scatT + 255) / 256), 256, 0, stream>>>(ei, xl, dinv, out, E);
}